// ContextHyperMatrix_3917010174732
// MI455X (gfx1250) — compile-verified
//
#include <hip/hip_runtime.h>

// ContextHyperMatrix: out[b,:] = x[b,:] @ weight[context[b]]
// Strategy: group rows by context (histogram + scan + scatter), then one
// workgroup per context does a small-M f32 WMMA GEMM with the 128x16
// B-operand slice held in registers across M-tiles.

#define IN_DIM 128
#define OUT_DIM 128
#define XS_STRIDE 132   // 16 rows * 132 floats: A-fragment ds_load_b64 hits all 64 banks

typedef __attribute__((ext_vector_type(2))) float v2f;
typedef __attribute__((ext_vector_type(8))) float v8f;

__global__ void zero_i32(int* __restrict__ p, int n) {
  int i = blockIdx.x * blockDim.x + threadIdx.x;
  if (i < n) p[i] = 0;
}

__global__ void hist_kernel(const int* __restrict__ ctx, int* __restrict__ counts, int B) {
  int i = blockIdx.x * blockDim.x + threadIdx.x;
  if (i < B) atomicAdd(&counts[ctx[i]], 1);
}

__global__ __launch_bounds__(1024) void scan_kernel(const int* __restrict__ counts,
                                                    int* __restrict__ offsets,
                                                    int* __restrict__ cursor, int C) {
  __shared__ int sm[1024];
  const int t = threadIdx.x;
  const int v = (t < C) ? counts[t] : 0;
  sm[t] = v;
  __syncthreads();
  for (int off = 1; off < 1024; off <<= 1) {
    int u = (t >= off) ? sm[t - off] : 0;
    __syncthreads();
    sm[t] += u;
    __syncthreads();
  }
  if (t < C) {
    offsets[t + 1] = sm[t];       // inclusive -> offsets[1..C]
    cursor[t] = sm[t] - v;        // exclusive prefix = running write cursor
    if (t == 0) offsets[0] = 0;
  }
}

__global__ void scatter_kernel(const int* __restrict__ ctx, int* __restrict__ cursor,
                               int* __restrict__ rowidx, int B) {
  int i = blockIdx.x * blockDim.x + threadIdx.x;
  if (i < B) {
    int c = ctx[i];
    int pos = atomicAdd(&cursor[c], 1);
    rowidx[pos] = i;
  }
}

// One workgroup per context c: out[rows_c, :] = x[rows_c, :] @ W_c (128x128 f32).
// 8 waves; wave w owns output columns [16w, 16w+16). B-operand (128x16 slice of
// W_c) lives in 64 VGPRs per lane, loaded once and reused for every M-tile.
__global__ __launch_bounds__(256) void ctx_gemm_wmma(
    const float* __restrict__ x, const float* __restrict__ weight,
    const int* __restrict__ offsets, const int* __restrict__ rowidx,
    float* __restrict__ out) {
  __shared__ float xs[16 * XS_STRIDE];

  const int c = blockIdx.x;
  const int start = offsets[c];
  const int cnt_all = offsets[c + 1] - start;
  if (cnt_all == 0) return;   // block-uniform

  const int tid = threadIdx.x;
  const int wave = tid >> 5;
  const int lane = tid & 31;
  const int laneHi = lane >> 4;    // 0 or 1
  const int laneLo = lane & 15;

  const float* W = weight + (size_t)c * (IN_DIM * OUT_DIM);
  const int n0 = wave * 16;

  // B-operand registers: breg[j] = W[k = 2*j + laneHi][n0 + laneLo]
  // (WMMA B layout: vgpr = k/2, lane = n + 16*(k%2); per K=4 step s the
  //  fragment is {breg[2s], breg[2s+1]}.)
  float breg[64];
  {
    const float* wp = W + laneHi * OUT_DIM + (n0 + laneLo);
#pragma unroll
    for (int j = 0; j < 64; ++j)
      breg[j] = wp[(size_t)(2 * j) * OUT_DIM];
  }

  const int ntiles = (cnt_all + 15) >> 4;
  for (int t = 0; t < ntiles; ++t) {
    const int base = start + (t << 4);
    const int tile_cnt = min(16, cnt_all - (t << 4));

    __syncthreads();   // previous tile's A reads done before overwrite
    // Cooperative gather of up to 16 x-rows into LDS (2 float4 per thread).
#pragma unroll
    for (int q = 0; q < 2; ++q) {
      int e = tid + q * 256;      // 0..511 float4 slots
      int m = e >> 5;             // row in tile
      int seg = e & 31;           // 16B segment within row
      float4 v = make_float4(0.f, 0.f, 0.f, 0.f);
      if (m < tile_cnt) {
        int row = rowidx[base + m];
        v = *(const float4*)(x + (size_t)row * IN_DIM + seg * 4);
      }
      *(float4*)(&xs[m * XS_STRIDE + seg * 4]) = v;
    }
    __syncthreads();

    // Accumulate 16x16 f32 tile over K=128 in 32 steps of K=4.
    v8f acc = {};
    // A layout 16x4: lanes 0-15 hold (K=0,K=1), lanes 16-31 hold (K=2,K=3).
    const float* xrow = &xs[laneLo * XS_STRIDE + laneHi * 2];
#pragma unroll
    for (int s = 0; s < 32; ++s) {
      v2f a = *(const v2f*)(xrow + s * 4);     // 8B-aligned ds_load_b64
      v2f b;
      b[0] = breg[2 * s];
      b[1] = breg[2 * s + 1];
      acc = __builtin_amdgcn_wmma_f32_16x16x4_f32(
          /*neg_a=*/false, a, /*neg_b=*/false, b,
          /*c_mod=*/(short)0, acc, /*reuse_a=*/false, /*reuse_b=*/false);
    }

    // C/D layout: VGPR r -> M = r + 8*laneHi, N = laneLo.
#pragma unroll
    for (int r = 0; r < 8; ++r) {
      int mr = r + 8 * laneHi;
      if (mr < tile_cnt) {
        int row = rowidx[base + mr];
        out[(size_t)row * OUT_DIM + (n0 + laneLo)] = acc[r];
      }
    }
  }
}

extern "C" void kernel_launch(void* const* d_in, const int* in_sizes, int n_in,
                              void* d_out, int out_size, void* d_ws, size_t ws_size,
                              hipStream_t stream) {
  const float* x = (const float*)d_in[0];
  const float* w = (const float*)d_in[1];
  const int* ctx = (const int*)d_in[2];
  float* out = (float*)d_out;

  const int B = in_sizes[0] / IN_DIM;                 // 32768
  const int C = in_sizes[1] / (IN_DIM * OUT_DIM);     // 1024

  // Workspace layout (ints): counts[C] | cursor[C] | offsets[C+1] | rowidx[B]
  int* ws = (int*)d_ws;
  int* counts = ws;
  int* cursor = ws + C;
  int* offsets = ws + 2 * C;
  int* rowidx = ws + 2 * C + (C + 1);

  zero_i32<<<(2 * C + 255) / 256, 256, 0, stream>>>(ws, 2 * C);
  hist_kernel<<<(B + 255) / 256, 256, 0, stream>>>(ctx, counts, B);
  scan_kernel<<<1, 1024, 0, stream>>>(counts, offsets, cursor, C);
  scatter_kernel<<<(B + 255) / 256, 256, 0, stream>>>(ctx, cursor, rowidx, B);
  ctx_gemm_wmma<<<C, 256, 0, stream>>>(x, w, offsets, rowidx, out);
}